// DualGraphRecurrentModule_44813688767170
// MI455X (gfx1250) — compile-verified
//
#include <hip/hip_runtime.h>
#include <math.h>

// ---------------- sizes (compile-time constants of the problem) -------------
#define NHEADS   4
#define DDIM     64
#define CDIM     64
#define HD       256          // NHEADS*DDIM
#define NEGSLOPE 0.2f

#define HASH_BITS 22
#define HASH_SIZE (1u << HASH_BITS)
#define HASH_MASK (HASH_SIZE - 1u)

typedef __attribute__((ext_vector_type(2))) float v2f;
typedef __attribute__((ext_vector_type(8))) float v8f;

// monotonic uint encoding of float for atomicMax-based segment_max
__device__ __forceinline__ unsigned enc_f(float f) {
    unsigned u = __float_as_uint(f);
    return (u & 0x80000000u) ? ~u : (u | 0x80000000u);
}
__device__ __forceinline__ float dec_f(unsigned u) {
    u = (u & 0x80000000u) ? (u & 0x7FFFFFFFu) : ~u;
    return __uint_as_float(u);
}

// ============================================================================
// WMMA GEMM: Y[n x Ncols] = X[n x 64] @ W^T   (W is Ncols x 64 row-major)
// One wave computes one 16x16 f32 tile via 16x V_WMMA_F32_16X16X4_F32.
// blockDim.x = 32 * (Ncols/16); gridDim.x = n/16  (n multiple of 16).
// A frag (16x4):  lanes 0-15 row=m0+lane K={k,k+1}; lanes 16-31 K={k+2,k+3}
// B frag (4x16):  lanes 0-15 col=n0+lane K={k,k+1}; lanes 16-31 K={k+2,k+3}
// C/D (16x16):    vgpr i: lanes 0-15 -> (m0+i, n0+lane); 16-31 -> (m0+i+8, ..)
// ============================================================================
__global__ void gat_wmma_gemm(const float* __restrict__ X,
                              const float* __restrict__ W,
                              const float* __restrict__ bias,
                              float* __restrict__ Y,
                              int Ncols)
{
    const int lane = threadIdx.x & 31;
    const int wave = threadIdx.x >> 5;
    const int m0   = blockIdx.x << 4;
    const int n0   = wave << 4;
    const int lr   = lane & 15;
    const int hi   = lane >> 4;

    const float* xrow = X + (size_t)(m0 + lr) * CDIM;
    const float* wrow = W + (size_t)(n0 + lr) * CDIM;

    v8f c = {};
#pragma unroll
    for (int k = 0; k < CDIM; k += 4) {
        const int ka = k + (hi << 1);
        v2f a = *(const v2f*)(xrow + ka);
        v2f b = *(const v2f*)(wrow + ka);
        c = __builtin_amdgcn_wmma_f32_16x16x4_f32(
                /*neg_a=*/false, a, /*neg_b=*/false, b,
                /*c_mod=*/(short)0, c, /*reuse_a=*/false, /*reuse_b=*/false);
    }

    const int col = n0 + lr;
    const float bv = bias ? bias[col] : 0.0f;
#pragma unroll
    for (int i = 0; i < 8; ++i) {
        const int row = m0 + i + (hi << 3);
        Y[(size_t)row * Ncols + col] = c[i] + bv;
    }
}

// ============================================================================
// Edge building: combine e1 | e2+n1 | e1+n1, drop self loops, dedup by hash.
// ============================================================================
__global__ void edge_insert(const long long* __restrict__ e1,
                            const long long* __restrict__ e2,
                            unsigned* __restrict__ table,
                            int* __restrict__ count,
                            int* __restrict__ esrc, int* __restrict__ edst,
                            int E, int n1, int n)
{
    const int i = blockIdx.x * blockDim.x + threadIdx.x;
    if (i >= 3 * E) return;
    int s, d;
    if (i < E)          { s = (int)e1[i];            d = (int)e1[E + i]; }
    else if (i < 2 * E) { const int j = i - E;
                          s = (int)e2[j] + n1;       d = (int)e2[E + j] + n1; }
    else                { const int j = i - 2 * E;
                          s = (int)e1[j] + n1;       d = (int)e1[E + j] + n1; }
    if (s == d) return;
    const unsigned key = (unsigned)s * (unsigned)n + (unsigned)d + 1u; // nonzero
    unsigned h = (key * 2654435761u) & HASH_MASK;
    for (;;) {
        const unsigned old = atomicCAS(&table[h], 0u, key);
        if (old == 0u) {
            const int pos = atomicAdd(count, 1);
            esrc[pos] = s; edst[pos] = d;
            return;
        }
        if (old == key) return;        // duplicate
        h = (h + 1u) & HASH_MASK;      // linear probe
    }
}

__global__ void add_self_loops(int* __restrict__ count,
                               int* __restrict__ esrc, int* __restrict__ edst,
                               int n)
{
    const int i = blockIdx.x * blockDim.x + threadIdx.x;
    if (i >= n) return;
    const int pos = atomicAdd(count, 1);
    esrc[pos] = i; edst[pos] = i;
}

// ============================================================================
// Per-(node,head): attention logits asrc/adst + init of m (segment max) and s.
// ============================================================================
__global__ void node_attn(const float* __restrict__ Hh,
                          const float* __restrict__ a_s,
                          const float* __restrict__ a_d,
                          float* __restrict__ asrc, float* __restrict__ adst,
                          unsigned* __restrict__ m_enc, float* __restrict__ ssum,
                          int n)
{
    const int i = blockIdx.x * blockDim.x + threadIdx.x;
    if (i >= n * NHEADS) return;
    const int head = i & (NHEADS - 1);
    const float* hp = Hh + (size_t)(i >> 2) * HD + head * DDIM;
    const float* as = a_s + head * DDIM;
    const float* ad = a_d + head * DDIM;
    float vs = 0.f, vd = 0.f;
#pragma unroll 8
    for (int d = 0; d < DDIM; ++d) { const float x = hp[d]; vs += x * as[d]; vd += x * ad[d]; }
    asrc[i] = vs; adst[i] = vd;
    m_enc[i] = 0u;      // identity for encoded-float max
    ssum[i]  = 0.f;
}

// ============================================================================
// Edge pass A: segment max of leaky_relu(asrc[src]+adst[dst]) over dst.
// ============================================================================
__global__ void edge_max(const int* __restrict__ esrc, const int* __restrict__ edst,
                         const int* __restrict__ count,
                         const float* __restrict__ asrc, const float* __restrict__ adst,
                         unsigned* __restrict__ m_enc)
{
    const int i = blockIdx.x * blockDim.x + threadIdx.x;
    const int e = i >> 2, head = i & 3;
    if (e >= *count) return;
    const int s = esrc[e], d = edst[e];
    float v = asrc[s * NHEADS + head] + adst[d * NHEADS + head];
    v = (v > 0.f) ? v : NEGSLOPE * v;
    atomicMax(&m_enc[d * NHEADS + head], enc_f(v));
}

// ============================================================================
// Edge pass B: ex = exp(e - m[dst]); s[dst] += ex; acc[dst] += ex * h[src].
// One wave per edge: lane = head*8 + j, each lane accumulates 8 floats.
// ============================================================================
__global__ void edge_accum(const int* __restrict__ esrc, const int* __restrict__ edst,
                           const int* __restrict__ count,
                           const float* __restrict__ asrc, const float* __restrict__ adst,
                           const unsigned* __restrict__ m_enc,
                           const float* __restrict__ Hh,
                           float* __restrict__ ssum, float* __restrict__ acc)
{
    const int gid  = blockIdx.x * blockDim.x + threadIdx.x;
    const int e    = gid >> 5;
    const int lane = gid & 31;
    if (e >= *count) return;
    const int head = lane >> 3;
    const int j    = lane & 7;
    const int s = esrc[e], d = edst[e];
    float v = asrc[s * NHEADS + head] + adst[d * NHEADS + head];
    v = (v > 0.f) ? v : NEGSLOPE * v;
    const float m  = dec_f(m_enc[d * NHEADS + head]);
    const float ex = expf(v - m);
    if (j == 0) atomicAdd(&ssum[d * NHEADS + head], ex);
    const float* hp = Hh  + (size_t)s * HD + head * DDIM + j * 8;
    float*       ap = acc + (size_t)d * HD + head * DDIM + j * 8;
#pragma unroll
    for (int t = 0; t < 8; ++t) atomicAdd(&ap[t], ex * hp[t]);
}

// ============================================================================
// Finalize layer: mean over heads of acc/(s+eps), + bias, optional relu.
// ============================================================================
__global__ void finalize_layer(const float* __restrict__ acc,
                               const float* __restrict__ ssum,
                               const float* __restrict__ bias,
                               float* __restrict__ hout,
                               int n, int do_relu)
{
    const int i = blockIdx.x * blockDim.x + threadIdx.x;
    if (i >= n * CDIM) return;
    const int node = i >> 6, c = i & 63;
    float v = 0.f;
#pragma unroll
    for (int h = 0; h < NHEADS; ++h)
        v += acc[(size_t)node * HD + h * DDIM + c] / (ssum[node * NHEADS + h] + 1e-16f);
    v = v * (1.0f / NHEADS) + bias[c];
    if (do_relu) v = fmaxf(v, 0.f);
    hout[i] = v;
}

// ============================================================================
// Final: rows n1..n1+n2 of h, L2-normalize, add residual.
// ============================================================================
__global__ void final_out(const float* __restrict__ hfin,
                          const float* __restrict__ resid,
                          float* __restrict__ out,
                          int n1, int n2)
{
    const int i = blockIdx.x * blockDim.x + threadIdx.x;
    if (i >= n2) return;
    const float* h2 = hfin + (size_t)(n1 + i) * CDIM;
    float ss = 0.f;
#pragma unroll 8
    for (int c = 0; c < CDIM; ++c) ss += h2[c] * h2[c];
    const float inv = 1.f / fmaxf(sqrtf(ss), 1e-12f);
    float* o = out + (size_t)i * CDIM;
    const float* r = resid + (size_t)i * CDIM;
#pragma unroll 8
    for (int c = 0; c < CDIM; ++c) o[c] = h2[c] * inv + r[c];
}

// ============================================================================
// Host-side orchestration
// ============================================================================
extern "C" void kernel_launch(void* const* d_in, const int* in_sizes, int n_in,
                              void* d_out, int out_size, void* d_ws, size_t ws_size,
                              hipStream_t stream)
{
    const float*     x1  = (const float*)d_in[0];
    const float*     x2  = (const float*)d_in[1];
    const long long* ei1 = (const long long*)d_in[2];
    const long long* ei2 = (const long long*)d_in[3];
    const float* Wl[4]  = { (const float*)d_in[4],  (const float*)d_in[8],
                            (const float*)d_in[12], (const float*)d_in[16] };
    const float* asl[4] = { (const float*)d_in[5],  (const float*)d_in[9],
                            (const float*)d_in[13], (const float*)d_in[17] };
    const float* adl[4] = { (const float*)d_in[6],  (const float*)d_in[10],
                            (const float*)d_in[14], (const float*)d_in[18] };
    const float* bl[4]  = { (const float*)d_in[7],  (const float*)d_in[11],
                            (const float*)d_in[15], (const float*)d_in[19] };
    const float* Wr = (const float*)d_in[20];
    const float* br = (const float*)d_in[21];
    float* out = (float*)d_out;

    const int n1 = in_sizes[0] / CDIM;
    const int n2 = in_sizes[1] / CDIM;
    const int n  = n1 + n2;
    const int E  = in_sizes[2] / 2;
    const int MAXE = 3 * E + n;

    // ---------------- workspace carve-up ----------------
    char* p = (char*)d_ws;
    auto take = [&](size_t bytes) { char* q = p; p += (bytes + 255) & ~size_t(255); return q; };
    unsigned* table  = (unsigned*)take((size_t)HASH_SIZE * 4);
    int*      esrc   = (int*)     take((size_t)MAXE * 4);
    int*      edst   = (int*)     take((size_t)MAXE * 4);
    int*      ecount = (int*)     take(256);
    float*    h_cur  = (float*)   take((size_t)n * CDIM * 4);
    float*    h_nxt  = (float*)   take((size_t)n * CDIM * 4);
    float*    Hh     = (float*)   take((size_t)n * HD * 4);
    float*    acc    = (float*)   take((size_t)n * HD * 4);
    float*    asrc   = (float*)   take((size_t)n * NHEADS * 4);
    float*    adst   = (float*)   take((size_t)n * NHEADS * 4);
    unsigned* m_enc  = (unsigned*)take((size_t)n * NHEADS * 4);
    float*    ssum   = (float*)   take((size_t)n * NHEADS * 4);
    float*    resid  = (float*)   take((size_t)n2 * CDIM * 4);
    (void)ws_size;

    // ---------------- build edge list (dedup + self loops) ----------------
    hipMemsetAsync(table, 0, (size_t)HASH_SIZE * 4, stream);
    hipMemsetAsync(ecount, 0, 4, stream);
    edge_insert<<<(3 * E + 255) / 256, 256, 0, stream>>>(ei1, ei2, table, ecount,
                                                         esrc, edst, E, n1, n);
    add_self_loops<<<(n + 255) / 256, 256, 0, stream>>>(ecount, esrc, edst, n);

    // ---------------- residual = x2 @ Wr^T + br ----------------
    gat_wmma_gemm<<<n2 / 16, 32 * (CDIM / 16), 0, stream>>>(x2, Wr, br, resid, CDIM);

    // ---------------- h = concat(x1, x2) ----------------
    hipMemcpyAsync(h_cur,                      x1, (size_t)n1 * CDIM * 4,
                   hipMemcpyDeviceToDevice, stream);
    hipMemcpyAsync(h_cur + (size_t)n1 * CDIM,  x2, (size_t)n2 * CDIM * 4,
                   hipMemcpyDeviceToDevice, stream);

    // ---------------- 4 GAT layers ----------------
    for (int layer = 0; layer < 4; ++layer) {
        const int is_final = (layer == 3);
        gat_wmma_gemm<<<n / 16, 32 * (HD / 16), 0, stream>>>(h_cur, Wl[layer],
                                                             nullptr, Hh, HD);
        node_attn<<<(n * NHEADS + 255) / 256, 256, 0, stream>>>(Hh, asl[layer], adl[layer],
                                                                asrc, adst, m_enc, ssum, n);
        hipMemsetAsync(acc, 0, (size_t)n * HD * 4, stream);
        edge_max<<<((size_t)MAXE * NHEADS + 255) / 256, 256, 0, stream>>>(
            esrc, edst, ecount, asrc, adst, m_enc);
        edge_accum<<<((size_t)MAXE * 32 + 255) / 256, 256, 0, stream>>>(
            esrc, edst, ecount, asrc, adst, m_enc, Hh, ssum, acc);
        finalize_layer<<<(n * CDIM + 255) / 256, 256, 0, stream>>>(
            acc, ssum, bl[layer], h_nxt, n, !is_final);
        float* t = h_cur; h_cur = h_nxt; h_nxt = t;
    }

    // ---------------- normalize + residual ----------------
    final_out<<<(n2 + 255) / 256, 256, 0, stream>>>(h_cur, resid, out, n1, n2);
}